// CandidateFinder_55929064128666
// MI455X (gfx1250) — compile-verified
//
#include <hip/hip_runtime.h>
#include <hip/hip_bf16.h>
#include <math.h>

#define B_    2
#define S_    4096
#define D_    12
#define KMAX  64
#define NH    4

typedef __attribute__((ext_vector_type(2))) float v2f;
typedef __attribute__((ext_vector_type(8))) float v8f;

__device__ __forceinline__ unsigned f2ord(float f) {
  unsigned u = __float_as_uint(f);
  return (u & 0x80000000u) ? ~u : (u | 0x80000000u);
}

// ---------------- Kernel 1: pack signature + LSH buckets into one u32 -------
// meta = sig[11:0] (sign pattern of batch B-1 row) | h0<<12 | h1<<17 | h2<<22 | h3<<27
__global__ void meta_kernel(const float* __restrict__ q,
                            const float* __restrict__ k,
                            const float* __restrict__ proj,
                            unsigned* __restrict__ qmeta,
                            unsigned* __restrict__ kmeta) {
  int idx = blockIdx.x * blockDim.x + threadIdx.x;
  if (idx >= B_ * S_) return;
  int s = idx & (S_ - 1);

  const float* qrow  = q + (size_t)idx * D_;
  const float* krow  = k + (size_t)idx * D_;
  const float* qlast = q + ((size_t)(B_ - 1) * S_ + s) * D_;
  const float* klast = k + ((size_t)(B_ - 1) * S_ + s) * D_;

  unsigned qm = 0, km = 0;
#pragma unroll
  for (int d = 0; d < D_; ++d) {
    qm |= (qlast[d] > 0.0f ? 1u : 0u) << d;
    km |= (klast[d] > 0.0f ? 1u : 0u) << d;
  }
#pragma unroll
  for (int j = 0; j < NH; ++j) {
    float aq = 0.f, ak = 0.f;
#pragma unroll
    for (int d = 0; d < D_; ++d) {
      float p = proj[d * NH + j];
      aq += qrow[d] * p;
      ak += krow[d] * p;
    }
    unsigned hq = (unsigned)(((int)floorf(aq * 0.25f)) & 31);
    unsigned hk = (unsigned)(((int)floorf(ak * 0.25f)) & 31);
    qm |= hq << (12 + 5 * j);
    km |= hk << (12 + 5 * j);
  }
  qmeta[idx] = qm;
  kmeta[idx] = km;
}

// Branchless candidate test: trie (12-bit signature equal) AND any of four
// 5-bit LSH fields equal. Exact zero-field detect for 5-bit fields.
__device__ __forceinline__ bool meta_match(unsigned qm, unsigned km) {
  const unsigned x = qm ^ km;
  const unsigned y = x >> 12;                              // four 5-bit fields
  const unsigned z = (y - 0x08421u) & ~y & 0x84210u;       // !=0 iff any field 0
  return ((x & 0xFFFu) == 0u) & (z != 0u);
}

// ---------------- Kernel 2: WMMA sims + mask + per-row selection ------------
__launch_bounds__(256)
__global__ void cand_kernel(const float* __restrict__ q,
                            const float* __restrict__ k,
                            const unsigned* __restrict__ qmeta,
                            const unsigned* __restrict__ kmeta,
                            int* __restrict__ out) {
  __shared__ float    sims[16][S_];      // 256 KB masked similarity panel
  __shared__ float    qrows[16][D_];
  __shared__ unsigned qmetaS[16];
  __shared__ int      topbuf[8][KMAX];

  const int tid  = threadIdx.x;
  const int lane = tid & 31;
  const int wave = tid >> 5;
  const int b     = blockIdx.x / (S_ / 16);
  const int qbase = (blockIdx.x % (S_ / 16)) * 16;

  if (tid < 16 * D_) {
    int r = tid / D_, d = tid - r * D_;
    qrows[r][d] = q[((size_t)b * S_ + qbase + r) * D_ + d];
  }
  if (tid < 16) qmetaS[tid] = qmeta[b * S_ + qbase + tid];
  __syncthreads();

  // A operand: 16x4 f32 tile. Lane l: M = l&15; VGPR0 holds K=(l<16?0:2), VGPR1 K+1.
  const int mrow = lane & 15;
  const int koff = (lane >> 4) * 2;
  v2f a0, a1, a2;
  a0.x = qrows[mrow][0 + koff]; a0.y = qrows[mrow][1 + koff];
  a1.x = qrows[mrow][4 + koff]; a1.y = qrows[mrow][5 + koff];
  a2.x = qrows[mrow][8 + koff]; a2.y = qrows[mrow][9 + koff];

  // Hoist this lane's 8 query-meta words (C rows m = mofs..mofs+7).
  const int mofs = (lane >> 4) << 3;
  unsigned qm8[8];
#pragma unroll
  for (int v = 0; v < 8; ++v) qm8[v] = qmetaS[mofs + v];

  // Phase 1: two adjacent 16-key tiles per iteration -> two independent
  // WMMA accumulation chains (ILP; occupancy is LDS-capped at 2 waves/SIMD).
  for (int jt = wave * 2; jt < S_ / 16; jt += 16) {
    const int kbase0 = jt * 16;
    const int kbase1 = kbase0 + 16;
    const float* krow0 = k + ((size_t)b * S_ + kbase0 + mrow) * D_;  // N = lane&15
    const float* krow1 = k + ((size_t)b * S_ + kbase1 + mrow) * D_;

    v2f p0, p1, p2, r0, r1, r2;
    p0.x = krow0[0 + koff]; p0.y = krow0[1 + koff];
    p1.x = krow0[4 + koff]; p1.y = krow0[5 + koff];
    p2.x = krow0[8 + koff]; p2.y = krow0[9 + koff];
    r0.x = krow1[0 + koff]; r0.y = krow1[1 + koff];
    r1.x = krow1[4 + koff]; r1.y = krow1[5 + koff];
    r2.x = krow1[8 + koff]; r2.y = krow1[9 + koff];
    const unsigned km0 = kmeta[b * S_ + kbase0 + mrow];
    const unsigned km1 = kmeta[b * S_ + kbase1 + mrow];

    v8f c0 = {}, c1 = {};
    c0 = __builtin_amdgcn_wmma_f32_16x16x4_f32(false, a0, false, p0, (short)0, c0, false, false);
    c1 = __builtin_amdgcn_wmma_f32_16x16x4_f32(false, a0, false, r0, (short)0, c1, false, false);
    c0 = __builtin_amdgcn_wmma_f32_16x16x4_f32(false, a1, false, p1, (short)0, c0, false, false);
    c1 = __builtin_amdgcn_wmma_f32_16x16x4_f32(false, a1, false, r1, (short)0, c1, false, false);
    c0 = __builtin_amdgcn_wmma_f32_16x16x4_f32(false, a2, false, p2, (short)0, c0, false, false);
    c1 = __builtin_amdgcn_wmma_f32_16x16x4_f32(false, a2, false, r2, (short)0, c1, false, false);

#pragma unroll
    for (int v = 0; v < 8; ++v) {
      const bool ok0 = meta_match(qm8[v], km0);            // branchless
      const bool ok1 = meta_match(qm8[v], km1);
      sims[mofs + v][kbase0 + mrow] = ok0 ? c0[v] : -INFINITY;
      sims[mofs + v][kbase1 + mrow] = ok1 ? c1[v] : -INFINITY;
    }
  }
  __syncthreads();

  // Phase 2: each wave owns 2 query rows; all selection is wave-local.
  volatile int* tb = topbuf[wave];
  for (int rr = 0; rr < 2; ++rr) {
    const int r = wave * 2 + rr;
    int* orow = out + ((size_t)b * S_ + qbase + r) * KMAX;

    // local scan: count + per-lane max (first/lowest index wins ties)
    float best = -INFINITY; int bestj = -1; int cnt = 0;
    for (int j = lane; j < S_; j += 32) {
      const float v = sims[r][j];
      if (v != -INFINITY) cnt++;
      if (v > best) { best = v; bestj = j; }
    }
    int total = cnt;
    for (int off = 16; off; off >>= 1) total += __shfl_xor(total, off);

    if (total <= KMAX) {
      // Branch B: leading -1 pads, then candidate indices ascending.
      const int base = KMAX - total;
      for (int i = lane; i < base; i += 32) orow[i] = -1;
      int running = 0;
      for (int cb = 0; cb < S_; cb += 32) {
        const float v = sims[r][cb + lane];
        const unsigned long long m = __ballot(v != -INFINITY);
        if (v != -INFINITY) {
          const int pos = running + __popcll(m & ((1ull << lane) - 1ull));
          orow[base + pos] = cb + lane;
        }
        running += __popcll(m);
      }
    } else {
      // Branch A: 64 rounds of wave argmax over 64-bit (sim, ~index) keys.
      for (int t = 0; t < KMAX; ++t) {
        const unsigned long long key =
            ((unsigned long long)f2ord(best) << 32) | (unsigned)(S_ - 1 - bestj);
        unsigned long long mk = key;
        for (int off = 16; off; off >>= 1) {
          const unsigned long long o = __shfl_xor(mk, off);
          if (o > mk) mk = o;
        }
        const int winj = (S_ - 1) - (int)(mk & 0xFFFFFFFFull);
        if (lane == 0) tb[t] = winj;
        if (key == mk) {  // unique winner: remove & rescan own stride set
          sims[r][bestj] = -INFINITY;
          best = -INFINITY; bestj = -1;
          for (int j = lane; j < S_; j += 32) {
            const float v = sims[r][j];
            if (v > best) { best = v; bestj = j; }
          }
        }
      }
      __builtin_amdgcn_wave_barrier();
      // rank-sort the 64 (distinct) indices ascending
#pragma unroll
      for (int e = lane; e < KMAX; e += 32) {
        const int val = tb[e];
        int rank = 0;
        for (int i = 0; i < KMAX; ++i) rank += (tb[i] < val) ? 1 : 0;
        orow[rank] = val;
      }
    }
  }
}

extern "C" void kernel_launch(void* const* d_in, const int* in_sizes, int n_in,
                              void* d_out, int out_size, void* d_ws, size_t ws_size,
                              hipStream_t stream) {
  const float* q    = (const float*)d_in[0];   // [B,S,D]
  const float* k    = (const float*)d_in[1];   // [B,S,D]
  const float* proj = (const float*)d_in[2];   // [D,NH]
  // d_in[3] = head_idx (unused by the reference math)

  unsigned* qmeta = (unsigned*)d_ws;
  unsigned* kmeta = qmeta + (size_t)B_ * S_;
  int* out = (int*)d_out;                      // [B,S,KMAX] int32

  meta_kernel<<<(B_ * S_ + 255) / 256, 256, 0, stream>>>(q, k, proj, qmeta, kmeta);
  cand_kernel<<<B_ * (S_ / 16), 256, 0, stream>>>(q, k, qmeta, kmeta, out);
}